// GraphNet_34462817583846
// MI455X (gfx1250) — compile-verified
//
#include <hip/hip_runtime.h>
#include <hip/hip_bf16.h>
#include <math.h>

#define NODES   100000
#define EDGES   800000
#define FDIM    64
#define NGRAPH  64
#define ROWTILES (NODES / 16)              // 6250
#define GEMM_BLOCKS ((ROWTILES + 7) / 8)   // 8 waves / block

typedef __attribute__((ext_vector_type(2))) float v2f;
typedef __attribute__((ext_vector_type(8))) float v8f;

// ---------------- degree / normalization ----------------
__global__ void init_deg_kernel(float* __restrict__ deg, int n) {
    int i = blockIdx.x * blockDim.x + threadIdx.x;
    if (i < n) deg[i] = 1.0f;              // self-loop
}

__global__ void count_deg_kernel(const int* __restrict__ dst,
                                 float* __restrict__ deg, int e) {
    int i = blockIdx.x * blockDim.x + threadIdx.x;
    if (i < e) atomicAdd(&deg[dst[i]], 1.0f);
}

__global__ void finish_dis_kernel(float* __restrict__ deg, int n) {
    int i = blockIdx.x * blockDim.x + threadIdx.x;
    if (i < n) deg[i] = rsqrtf(deg[i]);    // dis = 1/sqrt(deg)
}

// ---------------- dense projection: Out = (relu?)A @ W  via WMMA f32 16x16x4 ----------------
// RELU is a compile-time flag so the inner loop has no cndmask selects.
template <int RELU>
__global__ __launch_bounds__(256) void gemm_wmma_kernel(const float* __restrict__ A,
                                                        const float* __restrict__ W,
                                                        float* __restrict__ Out) {
    __shared__ float sW[FDIM * FDIM];      // 16 KB of 320 KB WGP LDS
    for (int i = threadIdx.x; i < FDIM * FDIM; i += blockDim.x) sW[i] = W[i];
    __syncthreads();

    const int wave  = threadIdx.x >> 5;    // 8 waves per block
    const int lane  = threadIdx.x & 31;
    const int rtile = blockIdx.x * 8 + wave;
    if (rtile >= ROWTILES) return;         // wave-uniform: EXEC all-1s inside
    const int m0    = rtile * 16;

    const int rhalf = lane >> 4;           // 0 | 1
    const int l15   = lane & 15;
    const int arow  = m0 + l15;

    v8f acc0 = {}, acc1 = {}, acc2 = {}, acc3 = {};
    #pragma unroll
    for (int k = 0; k < 16; ++k) {
        const int kbase = k * 4 + rhalf * 2;
        // A fragment: lane -> A[arow][kbase], A[arow][kbase+1]
        const float* ap = A + (size_t)arow * FDIM + kbase;
        v2f a; a.x = ap[0]; a.y = ap[1];
        if (RELU) { a.x = fmaxf(a.x, 0.f); a.y = fmaxf(a.y, 0.f); }
        // B fragments from LDS for the 4 column tiles (n0 = 0,16,32,48)
        const float* wp = sW + (size_t)kbase * FDIM + l15;
        v2f b0, b1, b2, b3;
        b0.x = wp[ 0]; b0.y = wp[ 64];
        b1.x = wp[16]; b1.y = wp[ 80];
        b2.x = wp[32]; b2.y = wp[ 96];
        b3.x = wp[48]; b3.y = wp[112];
        acc0 = __builtin_amdgcn_wmma_f32_16x16x4_f32(false, a, false, b0, (short)0, acc0, false, false);
        acc1 = __builtin_amdgcn_wmma_f32_16x16x4_f32(false, a, false, b1, (short)0, acc1, false, false);
        acc2 = __builtin_amdgcn_wmma_f32_16x16x4_f32(false, a, false, b2, (short)0, acc2, false, false);
        acc3 = __builtin_amdgcn_wmma_f32_16x16x4_f32(false, a, false, b3, (short)0, acc3, false, false);
    }
    // C/D layout: VGPR r -> row m0 + rhalf*8 + r, col n0 + l15
    float* op = Out + (size_t)(m0 + rhalf * 8) * FDIM + l15;
    #pragma unroll
    for (int r = 0; r < 8; ++r) {
        op[(size_t)r * FDIM +  0] = acc0[r];
        op[(size_t)r * FDIM + 16] = acc1[r];
        op[(size_t)r * FDIM + 32] = acc2[r];
        op[(size_t)r * FDIM + 48] = acc3[r];
    }
}

// ---------------- agg = dis^2 * hw + b (self-loop + bias) ----------------
__global__ void init_agg_kernel(const float* __restrict__ hw,
                                const float* __restrict__ dis,
                                const float* __restrict__ bias,
                                float* __restrict__ agg, int n_nodes) {
    int idx = blockIdx.x * blockDim.x + threadIdx.x;    // node*16 + quad
    int node = idx >> 4, q = idx & 15;
    if (node >= n_nodes) return;
    float d2 = dis[node]; d2 *= d2;
    float4 v  = ((const float4*)(hw + (size_t)node * FDIM))[q];
    float4 bb = ((const float4*)bias)[q];
    float4 r  = { v.x * d2 + bb.x, v.y * d2 + bb.y, v.z * d2 + bb.z, v.w * d2 + bb.w };
    ((float4*)(agg + (size_t)node * FDIM))[q] = r;
}

// ---------------- edge scatter: agg[dst] += dis[src]*dis[dst] * hw[src] ----------------
// one wave32 per edge; each lane handles a float2 of the 64 features (L2-resident atomics)
__global__ __launch_bounds__(256) void scatter_kernel(const float* __restrict__ hw,
                                                      const float* __restrict__ dis,
                                                      const int* __restrict__ src,
                                                      const int* __restrict__ dst,
                                                      float* __restrict__ agg, int n_edges) {
    int wid  = (blockIdx.x * blockDim.x + threadIdx.x) >> 5; // edge id (wave-uniform)
    int lane = threadIdx.x & 31;
    if (wid >= n_edges) return;
    int s = src[wid], d = dst[wid];
    float norm = dis[s] * dis[d];
    float2 v = ((const float2*)(hw + (size_t)s * FDIM))[lane];
    float* ap = agg + (size_t)d * FDIM + 2 * lane;
    atomicAdd(ap,     v.x * norm);
    atomicAdd(ap + 1, v.y * norm);
}

// ---------------- global max pool over sorted batch ----------------
__device__ inline void atomicMaxF(float* addr, float v) {
    if (v >= 0.0f) atomicMax((int*)addr, __float_as_int(v));
    else           atomicMin((unsigned int*)addr, __float_as_uint(v));
}

__global__ void pool_init_kernel(float* __restrict__ out, int n) {
    int i = blockIdx.x * blockDim.x + threadIdx.x;
    if (i < n) out[i] = -INFINITY;
}

__global__ __launch_bounds__(64) void pool_max_kernel(const float* __restrict__ h,
                                                      const int* __restrict__ batch,
                                                      float* __restrict__ out, int n_nodes) {
    int f  = threadIdx.x;                 // one thread per feature
    int n0 = blockIdx.x * 128;
    if (n0 >= n_nodes) return;
    int nend = min(n0 + 128, n_nodes);
    int cur = batch[n0];
    float lm = -INFINITY;
    for (int i = n0; i < nend; ++i) {
        int g = batch[i];                 // uniform across the block
        if (g != cur) {                   // batch is sorted: flush on boundary
            atomicMaxF(&out[(size_t)cur * FDIM + f], lm);
            cur = g; lm = -INFINITY;
        }
        lm = fmaxf(lm, h[(size_t)i * FDIM + f]);
    }
    atomicMaxF(&out[(size_t)cur * FDIM + f], lm);
}

// ---------------- driver ----------------
extern "C" void kernel_launch(void* const* d_in, const int* in_sizes, int n_in,
                              void* d_out, int out_size, void* d_ws, size_t ws_size,
                              hipStream_t stream) {
    const float* x     = (const float*)d_in[0];
    const int*   eidx  = (const int*)d_in[1];
    const int*   batch = (const int*)d_in[2];
    const float* W1 = (const float*)d_in[3];
    const float* b1 = (const float*)d_in[4];
    const float* W2 = (const float*)d_in[5];
    const float* b2 = (const float*)d_in[6];
    const float* W3 = (const float*)d_in[7];
    const float* b3 = (const float*)d_in[8];
    const int* src = eidx;
    const int* dst = eidx + EDGES;
    float* out = (float*)d_out;

    // workspace carve-out: dis | hw | agg  (~51.6 MB)
    float* dis = (float*)d_ws;
    float* hw  = dis + 100352;                      // NODES padded to 256-float align
    float* agg = hw + (size_t)NODES * FDIM;

    const int TPB = 256;
    dim3 nblk((NODES + TPB - 1) / TPB);
    dim3 eblk((EDGES + TPB - 1) / TPB);
    dim3 iblk((NODES * 16 + TPB - 1) / TPB);
    dim3 sblk(((size_t)EDGES * 32 + TPB - 1) / TPB);

    // normalization: dis = rsqrt(1 + in-degree)
    init_deg_kernel<<<nblk, TPB, 0, stream>>>(dis, NODES);
    count_deg_kernel<<<eblk, TPB, 0, stream>>>(dst, dis, EDGES);
    finish_dis_kernel<<<nblk, TPB, 0, stream>>>(dis, NODES);

    // layer 1 (input x, no input relu)
    gemm_wmma_kernel<0><<<GEMM_BLOCKS, TPB, 0, stream>>>(x, W1, hw);
    init_agg_kernel<<<iblk, TPB, 0, stream>>>(hw, dis, b1, agg, NODES);
    scatter_kernel<<<sblk, TPB, 0, stream>>>(hw, dis, src, dst, agg, EDGES);

    // layer 2 (relu fused into GEMM A-load)
    gemm_wmma_kernel<1><<<GEMM_BLOCKS, TPB, 0, stream>>>(agg, W2, hw);
    init_agg_kernel<<<iblk, TPB, 0, stream>>>(hw, dis, b2, agg, NODES);
    scatter_kernel<<<sblk, TPB, 0, stream>>>(hw, dis, src, dst, agg, EDGES);

    // layer 3 (relu fused, no output relu)
    gemm_wmma_kernel<1><<<GEMM_BLOCKS, TPB, 0, stream>>>(agg, W3, hw);
    init_agg_kernel<<<iblk, TPB, 0, stream>>>(hw, dis, b3, agg, NODES);
    scatter_kernel<<<sblk, TPB, 0, stream>>>(hw, dis, src, dst, agg, EDGES);

    // global max pool
    pool_init_kernel<<<(NGRAPH * FDIM + TPB - 1) / TPB, TPB, 0, stream>>>(out, NGRAPH * FDIM);
    pool_max_kernel<<<(NODES + 127) / 128, 64, 0, stream>>>(agg, batch, out, NODES);
}